// CTCLoss_23751169147343
// MI455X (gfx1250) — compile-verified
//
#include <hip/hip_runtime.h>
#include <hip/hip_bf16.h>

#define NEGV (-1.0e30f)
#define PIPE_D 8                 // prefetch pipeline depth (rows in flight)
#define WAIT_STEADY (2 * (PIPE_D - 1))  // asynccnt target in steady state = 14

// ---------------------------------------------------------------------------
// CDNA5 async global->LDS path (ASYNCcnt). Guarded so the file always compiles;
// fallback is a synchronous VGPR-staged copy.
// ---------------------------------------------------------------------------
#if defined(__has_builtin)
#if __has_builtin(__builtin_amdgcn_global_load_async_to_lds_b128) && \
    __has_builtin(__builtin_amdgcn_s_wait_asynccnt)
#define CTC_ASYNC 1
#endif
#endif

typedef int v4i __attribute__((vector_size(16)));
typedef __attribute__((address_space(1))) v4i* g_v4i_p;  // global (device) int4*
typedef __attribute__((address_space(3))) v4i* l_v4i_p;  // LDS int4*

// Copy one 256-float (1 KB) row from global to LDS. One wave32: each lane moves
// 16B twice -> two GLOBAL_LOAD_ASYNC_TO_LDS_B128 per row (2 ASYNCcnt ticks).
__device__ __forceinline__ void prefetch_row(const float* __restrict__ g,
                                             float* l, int lane) {
#if CTC_ASYNC
  g_v4i_p gp0 = (g_v4i_p)(g + lane * 4);
  l_v4i_p lp0 = (l_v4i_p)(l + lane * 4);
  __builtin_amdgcn_global_load_async_to_lds_b128(gp0, lp0, 0, 0);
  g_v4i_p gp1 = (g_v4i_p)(g + 128 + lane * 4);
  l_v4i_p lp1 = (l_v4i_p)(l + 128 + lane * 4);
  __builtin_amdgcn_global_load_async_to_lds_b128(gp1, lp1, 0, 0);
#else
  float4 v0 = ((const float4*)g)[lane];
  float4 v1 = ((const float4*)g)[lane + 32];
  ((float4*)l)[lane] = v0;
  ((float4*)l)[lane + 32] = v1;
#endif
}

__device__ __forceinline__ void wait_async_steady() {
#if CTC_ASYNC
  __builtin_amdgcn_s_wait_asynccnt(WAIT_STEADY);
#endif
}

// ---------------------------------------------------------------------------
// log-sum-exp helpers matching the reference's NEG masking semantics.
// __expf/__logf lower to single v_exp_f32 / v_log_f32 (TRANS, co-executes).
// ---------------------------------------------------------------------------
__device__ __forceinline__ float lse3(float x, float y, float z) {
  float m = fmaxf(fmaxf(x, y), z);
  bool valid = m > -5.0e29f;
  float ms = valid ? m : 0.0f;
  float s = __expf(x - ms) + __expf(y - ms) + __expf(z - ms);
  return valid ? (ms + __logf(s)) : NEGV;
}

__device__ __forceinline__ float lse2(float x, float y) {
  float m = fmaxf(x, y);
  bool valid = m > -5.0e29f;
  float ms = valid ? m : 0.0f;
  float s = __expf(x - ms) + __expf(y - ms);
  return valid ? (ms + __logf(s)) : NEGV;
}

// ---------------------------------------------------------------------------
// Kernel 1: one wave32 per batch element. 5 alpha states per lane (L<=160).
// Recurrence lives in registers; neighbor states come via __shfl_up; emit
// values are gathered from an 8-deep async-pipelined LDS row buffer.
// Slot assignment: row r lives in buf[r % PIPE_D]; each slot has exactly one
// async writer per period, and gathers of a slot complete (dscnt-waited at
// their use) before the slot's next writer is issued one iteration later.
// Row 0 is never staged in LDS: alpha0 is initialized straight from global,
// so no ds_load races the first async writes.
// ---------------------------------------------------------------------------
__global__ void __launch_bounds__(32)
ctc_alpha_kernel(const float* __restrict__ lp,   // [T,B,C]
                 const int* __restrict__ tgt,    // [B,S]
                 const int* __restrict__ ilen,   // [B]
                 const int* __restrict__ tlen,   // [B]
                 int T, int B, int C, int S,
                 float* __restrict__ ws)         // [B] per-b normalized loss
{
  __shared__ float buf[PIPE_D][256];  // C == 256, 8-deep pipeline (8 KB)

  const int b = blockIdx.x;
  const int lane = threadIdx.x & 31;
  const int L = 2 * S + 1;
  const size_t strideT = (size_t)B * (size_t)C;
  const float* rowBase = lp + (size_t)b * (size_t)C;

  // Per-lane labels and skip_ok for states l = 5*lane + k.
  int lab[5];
  bool skip[5];
#pragma unroll
  for (int k = 0; k < 5; ++k) {
    int l = lane * 5 + k;
    int labv = 0;
    bool sk = false;
    if (l < L && (l & 1)) {
      int j = (l - 1) >> 1;
      labv = tgt[b * S + j];
      if (j >= 1) {
        int prev = tgt[b * S + j - 1];
        sk = (labv != 0) && (labv != prev);
      }
    }
    lab[k] = labv;
    skip[k] = sk;
  }

  // Prime the pipeline: rows 1..PIPE_D-1 into slots 1..PIPE_D-1.
#pragma unroll
  for (int i = 1; i < PIPE_D; ++i) {
    int r = (i < T) ? i : (T - 1);
    prefetch_row(rowBase + (size_t)r * strideT, &buf[i][0], lane);
  }

  // alpha0 straight from global (row 0 is never staged in LDS).
  float a[5];
#pragma unroll
  for (int k = 0; k < 5; ++k) a[k] = NEGV;
  if (lane == 0) {
    a[0] = rowBase[0];       // emit[0][b][blank]
    a[1] = rowBase[lab[1]];  // emit[0][b][ext[1]]
  }

  const int myIlen = ilen[b];

  for (int t = 1; t < T; ++t) {
    // Issue row t+PIPE_D-1 (clamped) into slot (t-1) % PIPE_D, then wait so
    // that row t (the oldest in flight) has fully landed in LDS.
    int tn = (t + PIPE_D - 1 < T) ? (t + PIPE_D - 1) : (T - 1);
    prefetch_row(rowBase + (size_t)tn * strideT,
                 &buf[(t + PIPE_D - 1) & (PIPE_D - 1)][0], lane);
    wait_async_steady();  // ASYNCcnt <= 2*(PIPE_D-1): row t complete

    const float* r = &buf[t & (PIPE_D - 1)][0];
    float e[5];
#pragma unroll
    for (int k = 0; k < 5; ++k) e[k] = r[lab[k]];  // ds_load_b32 gather

    float p4 = __shfl_up(a[4], 1);  // alpha[l-1] for k==0
    float p3 = __shfl_up(a[3], 1);  // alpha[l-2] for k==0
    if (lane == 0) { p4 = NEGV; p3 = NEGV; }

    float n[5];
    n[0] = lse3(a[0], p4,   skip[0] ? p3   : NEGV) + e[0];
    n[1] = lse3(a[1], a[0], skip[1] ? p4   : NEGV) + e[1];
    n[2] = lse3(a[2], a[1], skip[2] ? a[0] : NEGV) + e[2];
    n[3] = lse3(a[3], a[2], skip[3] ? a[1] : NEGV) + e[3];
    n[4] = lse3(a[4], a[3], skip[4] ? a[2] : NEGV) + e[4];

    bool act = (t < myIlen);
#pragma unroll
    for (int k = 0; k < 5; ++k) a[k] = act ? n[k] : a[k];
  }

  // Final: ll = lse2(alpha[2*tl], alpha[2*tl-1]); loss = -ll (masked) / tl.
  int tl = tlen[b];
  int hi = 2 * tl;
  int lo = hi - 1;
  int laneHi = hi / 5, kHi = hi - laneHi * 5;
  int laneLo = lo / 5, kLo = lo - laneLo * 5;

  float vh[5], vl[5];
#pragma unroll
  for (int k = 0; k < 5; ++k) {
    vh[k] = __shfl(a[k], laneHi);
    vl[k] = __shfl(a[k], laneLo);
  }
  float ah = (kHi == 0) ? vh[0] : (kHi == 1) ? vh[1] : (kHi == 2) ? vh[2]
           : (kHi == 3) ? vh[3] : vh[4];
  float al = (kLo == 0) ? vl[0] : (kLo == 1) ? vl[1] : (kLo == 2) ? vl[2]
           : (kLo == 3) ? vl[3] : vl[4];

  float ll = lse2(ah, al);
  float loss = (ll > -5.0e29f) ? -ll : 0.0f;
  if (lane == 0) ws[b] = loss / (float)tl;
}

// ---------------------------------------------------------------------------
// Kernel 2: deterministic mean reduction of ws[0..B) -> out[0].
// ---------------------------------------------------------------------------
__global__ void __launch_bounds__(128)
ctc_reduce_kernel(const float* __restrict__ ws, float* __restrict__ out, int B) {
  __shared__ float s[128];
  int i = threadIdx.x;
  float v = 0.0f;
  for (int j = i; j < B; j += 128) v += ws[j];
  s[i] = v;
  __syncthreads();
  for (int off = 64; off > 0; off >>= 1) {
    if (i < off) s[i] += s[i + off];
    __syncthreads();
  }
  if (i == 0) out[0] = s[0] / (float)B;
}

extern "C" void kernel_launch(void* const* d_in, const int* in_sizes, int n_in,
                              void* d_out, int out_size, void* d_ws, size_t ws_size,
                              hipStream_t stream) {
  const float* log_probs = (const float*)d_in[0];
  const int* targets     = (const int*)d_in[1];
  const int* input_len   = (const int*)d_in[2];
  const int* target_len  = (const int*)d_in[3];

  const int B = in_sizes[2];           // 128
  const int S = in_sizes[1] / B;       // 64
  const int C = 256;                   // reference num classes (LDS buffer size)
  const int T = (int)((long long)in_sizes[0] / ((long long)B * C));  // 1024

  float* ws = (float*)d_ws;            // B floats of scratch

  ctc_alpha_kernel<<<B, 32, 0, stream>>>(log_probs, targets, input_len,
                                         target_len, T, B, C, S, ws);
  ctc_reduce_kernel<<<1, 128, 0, stream>>>(ws, (float*)d_out, B);
}